// SoftVQLayer_28046136443277
// MI455X (gfx1250) — compile-verified
//
#include <hip/hip_runtime.h>
#include <hip/hip_bf16.h>

// ---------------------------------------------------------------------------
// SoftVQ fused kernel for MI455X (gfx1250, wave32).
//   B=4 L=2048 D=1024 V=32 K=4096  -> 8192 tokens.
//
// v3: same 4-waves-per-tile cooperative scheme as v2 (512 blocks x 128
// threads = 2048 waves), but the wave id is pinned to an SGPR via
// readfirstlane so all per-wave loop bounds and LDS bases are scalar:
// loop control becomes s_cmp/s_cbranch instead of v_cmp + exec masking
// (which also keeps EXEC untouched around the WMMA loops, as the ISA
// requires).
//   - Stage 1 (hp = h @ Wp^T): split-K across waves, partials merged in LDS.
//   - Stages 2+3 (logits + online softmax + h_vq accum): codebook chunks
//     round-robined across waves; per-wave flash-softmax state merged with
//     exp(2*(m_w - M)) rescaling at the end.
//   - Stage 4 (out = h_vq @ Wpi^T): N split across waves.
// All matrix math uses V_WMMA_F32_16X16X4_F32. A[8192,4096] (134 MB) never
// touches memory; HBM traffic ~67 MB -> ~2.9us floor at 23.3 TB/s.
// ---------------------------------------------------------------------------

typedef __attribute__((ext_vector_type(2))) float v2f;
typedef __attribute__((ext_vector_type(8))) float v8f;

__device__ __forceinline__ v8f wmma_f32(v2f a, v2f b, v8f c) {
    // D = A(16x4) * B(4x16) + C(16x16), fp32.
    return __builtin_amdgcn_wmma_f32_16x16x4_f32(
        false, a, false, b, (short)0, c, false, false);
}

#define LDS_WAIT() asm volatile("s_wait_dscnt 0x0" ::: "memory")

constexpr int Bc = 4, Lc = 2048, Dc = 1024, Vc = 32, Kc = 4096;
constexpr int ROWS = Bc * Lc;   // 8192
constexpr int TILE = 16;
constexpr int NW   = 4;         // waves per block (cooperating on one tile)

__global__ __launch_bounds__(32 * NW) void softvq_fused_kernel(
    const float* __restrict__ h,     // [ROWS, D]
    const float* __restrict__ amask, // [ROWS]
    const float* __restrict__ Wp,    // [V, D]
    const float* __restrict__ bp,    // [V]
    const float* __restrict__ Wpi,   // [D, V]
    const float* __restrict__ bpi,   // [D]
    const float* __restrict__ emb,   // [K, V]
    float* __restrict__ out_q,       // [ROWS, D]
    float* __restrict__ out_code,    // [ROWS] (codes as float)
    float* __restrict__ out_loss)    // [1]
{
    // Padded strides keep 8B alignment for b64 LDS ops and avoid bank
    // conflicts on the stride-per-lane transpose reads.
    __shared__ float hp_s[16][34];        // merged hp_n / h_vq (A staging)
    __shared__ float es[NW][16][34];      // per-wave normalized emb chunk
    __shared__ float ps[NW][16][18];      // per-wave softmax p tile
    __shared__ float part[NW][16][33];    // split-K partials / per-wave h_vq
    __shared__ float mst[NW][16], lst[NW][16], bvs[NW][16];
    __shared__ int   bis[NW][16];

    const int tid  = threadIdx.x;
    // Wave id is uniform across the wave: pin it in an SGPR so loop bounds
    // and per-wave LDS bases are scalar (s_cmp loop control, no exec churn).
    const int wid  = __builtin_amdgcn_readfirstlane(tid >> 5);   // 0..3
    const int lane = tid & 31;
    const int hi   = lane >> 4;      // half-wave id
    const int li   = lane & 15;
    const int kd   = hi * 2;         // K-pair offset inside a 16x16x4 fragment
    const int row0 = blockIdx.x * TILE;

    // =====================================================================
    // Stage 1 (split-K): wave w reduces k in [w*256, w*256+256).
    // C-layout: vgpr r holds row (r + 8*hi); acc0 col=li, acc1 col=li+16.
    // =====================================================================
    v8f acc0 = {}, acc1 = {};
    {
        const float* hrow = h + (size_t)(row0 + li) * Dc;   // A: M = li
        const float* w0   = Wp + (size_t)li * Dc;           // B tile0: n = li
        const float* w1   = Wp + (size_t)(li + 16) * Dc;    // B tile1: n = li+16
        const int kbeg = wid * (Dc / NW), kend = kbeg + (Dc / NW);
        for (int k0 = kbeg; k0 < kend; k0 += 4) {
            v2f a, b0, b1;
            a.x  = hrow[k0 + kd]; a.y  = hrow[k0 + kd + 1];
            b0.x = w0[k0 + kd];   b0.y = w0[k0 + kd + 1];
            b1.x = w1[k0 + kd];   b1.y = w1[k0 + kd + 1];
            acc0 = wmma_f32(a, b0, acc0);
            acc1 = wmma_f32(a, b1, acc1);
        }
    }
#pragma unroll
    for (int r = 0; r < 8; ++r) {
        part[wid][r + 8 * hi][li]      = acc0[r];
        part[wid][r + 8 * hi][li + 16] = acc1[r];
    }
    __syncthreads();

    // Sum split-K partials, add bias, L2-normalize rows (redundant per wave).
    {
        const float bp0 = bp[li], bp1 = bp[li + 16];
#pragma unroll
        for (int r = 0; r < 8; ++r) {
            const int row = r + 8 * hi;
            float x0 = bp0, x1 = bp1;
#pragma unroll
            for (int w = 0; w < NW; ++w) {
                x0 += part[w][row][li];
                x1 += part[w][row][li + 16];
            }
            float t = x0 * x0 + x1 * x1;
            t += __shfl_xor(t, 1, 32); t += __shfl_xor(t, 2, 32);
            t += __shfl_xor(t, 4, 32); t += __shfl_xor(t, 8, 32);
            float rn = rsqrtf(t);
            acc0[r] = x0 * rn; acc1[r] = x1 * rn;
        }
    }
    if (wid == 0) {
#pragma unroll
        for (int r = 0; r < 8; ++r) {
            hp_s[r + 8 * hi][li]      = acc0[r];
            hp_s[r + 8 * hi][li + 16] = acc1[r];
        }
    }
    __syncthreads();

    v2f av[8];                               // hp_n A fragments (all chunks)
#pragma unroll
    for (int j = 0; j < 8; ++j) {
        av[j].x = hp_s[li][4 * j + kd];
        av[j].y = hp_s[li][4 * j + kd + 1];
    }

    // =====================================================================
    // Stage 2+3: wave w streams codebook chunks kc = w*16, w*16+64, ...
    // Online softmax of logits 2*s (since -d = 2s - 2) + WMMA h_vq accum.
    // =====================================================================
    float mrow[8], lrow[8], bestv[8];
    int   besti[8];
#pragma unroll
    for (int r = 0; r < 8; ++r) {
        mrow[r] = -3.0e38f; lrow[r] = 0.f; bestv[r] = -3.0e38f; besti[r] = 0;
    }
    v8f hv0 = {}, hv1 = {};                  // per-wave h_vq partials

    for (int kc = wid * 16; kc < Kc; kc += 16 * NW) {
        // Load + normalize 16 emb rows; these registers ARE the B fragments
        // for the s-tile WMMAs (lane n = emb row kc+li, K = v dimension).
        const float* er = emb + (size_t)(kc + li) * Vc;
        v2f eb[8];
        float ss = 0.f;
#pragma unroll
        for (int j = 0; j < 8; ++j) {
            eb[j].x = er[4 * j + kd]; eb[j].y = er[4 * j + kd + 1];
            ss += eb[j].x * eb[j].x + eb[j].y * eb[j].y;
        }
        ss += __shfl_xor(ss, 16, 32);        // combine the two half-rows
        const float rn = rsqrtf(ss);

        v8f sc = {};
#pragma unroll
        for (int j = 0; j < 8; ++j) {
            eb[j].x *= rn; eb[j].y *= rn;
            *(v2f*)&es[wid][li][4 * j + kd] = eb[j];  // stage normalized chunk
            sc = wmma_f32(av[j], eb[j], sc);          // s[m][kc+li]
        }

        // Online softmax update in C-layout (row = r+8*hi, col = kc+li).
#pragma unroll
        for (int r = 0; r < 8; ++r) {
            float v = sc[r];
            if (v > bestv[r]) { bestv[r] = v; besti[r] = kc + li; }
            float cm = v;
            cm = fmaxf(cm, __shfl_xor(cm, 1, 32));
            cm = fmaxf(cm, __shfl_xor(cm, 2, 32));
            cm = fmaxf(cm, __shfl_xor(cm, 4, 32));
            cm = fmaxf(cm, __shfl_xor(cm, 8, 32));
            float nm = fmaxf(mrow[r], cm);
            float sf = __expf(2.f * (mrow[r] - nm));    // rescale old state
            float p  = __expf(2.f * (v - nm));
            float psum = p;
            psum += __shfl_xor(psum, 1, 32); psum += __shfl_xor(psum, 2, 32);
            psum += __shfl_xor(psum, 4, 32); psum += __shfl_xor(psum, 8, 32);
            lrow[r] = lrow[r] * sf + psum;
            mrow[r] = nm;
            hv0[r] *= sf; hv1[r] *= sf;
            ps[wid][r + 8 * hi][li] = p;                // p tile for stage 3
        }
        LDS_WAIT();   // wave-private buffers: in-order DS handoff suffices

        // Stage 3: hv[16,32] += p[16,16] @ emb_n_chunk[16,32].
#pragma unroll
        for (int j = 0; j < 4; ++j) {
            v2f ap, b0, b1;
            ap.x = ps[wid][li][4 * j + kd];
            ap.y = ps[wid][li][4 * j + kd + 1];
            b0.x = es[wid][4 * j + kd][li];
            b0.y = es[wid][4 * j + kd + 1][li];
            b1.x = es[wid][4 * j + kd][li + 16];
            b1.y = es[wid][4 * j + kd + 1][li + 16];
            hv0 = wmma_f32(ap, b0, hv0);
            hv1 = wmma_f32(ap, b1, hv1);
        }
    }

    // Publish per-wave softmax state for the cross-wave merge.
#pragma unroll
    for (int r = 0; r < 8; ++r) {
        const int row = r + 8 * hi;
        part[wid][row][li]      = hv0[r];
        part[wid][row][li + 16] = hv1[r];
    }
    // Per-wave argmax across the 16 lanes of each half-wave, then publish.
#pragma unroll
    for (int r = 0; r < 8; ++r) {
        float bv = bestv[r]; int bi = besti[r];
#pragma unroll
        for (int m = 1; m < 16; m <<= 1) {
            float ov = __shfl_xor(bv, m, 32);
            int   oi = __shfl_xor(bi, m, 32);
            if (ov > bv || (ov == bv && oi < bi)) { bv = ov; bi = oi; }
        }
        if (li == 0) {
            const int row = r + 8 * hi;
            mst[wid][row] = mrow[r];
            lst[wid][row] = lrow[r];
            bvs[wid][row] = bv;
            bis[wid][row] = bi;
        }
    }
    __syncthreads();

    // =====================================================================
    // Merge the NW partial softmax states: M = max m_w,
    // L = sum l_w * e^{2(m_w-M)}, h_vq = sum hv_w * e^{2(m_w-M)}; then
    // A-normalize (/L), apply mask, stage into hp_s for stage 4.
    // =====================================================================
    if (wid == 0) {
#pragma unroll
        for (int r = 0; r < 8; ++r) {
            const int row = r + 8 * hi;
            float M = mst[0][row];
#pragma unroll
            for (int w = 1; w < NW; ++w) M = fmaxf(M, mst[w][row]);
            float L = 0.f, y0 = 0.f, y1 = 0.f;
#pragma unroll
            for (int w = 0; w < NW; ++w) {
                float sf = __expf(2.f * (mst[w][row] - M));
                L  += lst[w][row] * sf;
                y0 += part[w][row][li]      * sf;
                y1 += part[w][row][li + 16] * sf;
            }
            float mk  = amask[row0 + row];
            float inv = (mk == 1.f ? 1.f : 0.f) / L;
            hp_s[row][li]      = y0 * inv;
            hp_s[row][li + 16] = y1 * inv;
        }
    }
    // Cross-wave argmax merge + code write (threads 0..15 of wave 0).
    if (tid < 16) {
        const int row = tid;
        float bv = bvs[0][row]; int bi = bis[0][row];
#pragma unroll
        for (int w = 1; w < NW; ++w) {
            float ov = bvs[w][row]; int oi = bis[w][row];
            if (ov > bv || (ov == bv && oi < bi)) { bv = ov; bi = oi; }
        }
        float mk = amask[row0 + row];
        out_code[row0 + row] = (mk == 1.f) ? (float)bi : 0.f;
    }
    __syncthreads();

    // =====================================================================
    // Stage 4: out[16,1024] = h_vq[16,32] @ Wpi^T + bpi; N split across waves.
    // =====================================================================
#pragma unroll
    for (int j = 0; j < 8; ++j) {
        av[j].x = hp_s[li][4 * j + kd];
        av[j].y = hp_s[li][4 * j + kd + 1];
    }
    for (int nt = wid * 16; nt < Dc; nt += 16 * NW) {
        v8f oc = {};
        const float* wrow = Wpi + (size_t)(nt + li) * Vc;  // B[v][n]=Wpi[n][v]
#pragma unroll
        for (int j = 0; j < 8; ++j) {
            v2f b; b.x = wrow[4 * j + kd]; b.y = wrow[4 * j + kd + 1];
            oc = wmma_f32(av[j], b, oc);
        }
        const float bo = bpi[nt + li];
        float* op = out_q + (size_t)row0 * Dc + nt + li;
#pragma unroll
        for (int r = 0; r < 8; ++r)
            op[(size_t)(r + 8 * hi) * Dc] = oc[r] + bo;
    }

    if (blockIdx.x == 0 && threadIdx.x == 0) out_loss[0] = 0.f;
}

// ---------------------------------------------------------------------------
extern "C" void kernel_launch(void* const* d_in, const int* in_sizes, int n_in,
                              void* d_out, int out_size, void* d_ws, size_t ws_size,
                              hipStream_t stream) {
    const float* h    = (const float*)d_in[0];
    const float* mask = (const float*)d_in[1];
    const float* Wp   = (const float*)d_in[2];
    const float* bp   = (const float*)d_in[3];
    const float* Wpi  = (const float*)d_in[4];
    const float* bpi  = (const float*)d_in[5];
    const float* emb  = (const float*)d_in[6];

    float* out      = (float*)d_out;
    float* out_q    = out;                               // [ROWS*D]
    float* out_code = out + (size_t)ROWS * Dc;           // [ROWS]
    float* out_loss = out_code + ROWS;                   // [1]

    softvq_fused_kernel<<<ROWS / TILE, 32 * NW, 0, stream>>>(
        h, mask, Wp, bp, Wpi, bpi, emb, out_q, out_code, out_loss);
}